// AttentionDeLELSTM_11235634446893
// MI455X (gfx1250) — compile-verified
//
#include <hip/hip_runtime.h>
#include <math.h>

// ---------------- problem constants ----------------
constexpr int   BATCH  = 64;
constexpr int   SEQT   = 48;
constexpr int   DIMD   = 16;
constexpr int   HID    = 128;
constexpr int   NUU    = 128;
constexpr int   NHEAD  = 8;
constexpr int   KK     = 272;          // 2*D + D*(D-1)
constexpr int   KP     = 320;          // KK padded to multiple of 64
constexpr int   TSTEPS = 47;           // T-1
constexpr int   TOFF   = 3;            // SHORT+1
constexpr float THRv   = 0.01f;
constexpr float LAMv   = 0.01f;

constexpr size_t SZ_HV = (size_t)BATCH * DIMD * HID;   // 131072

// workspace layout (floats)
constexpr size_t OFF_HVAR  = 0;
constexpr size_t OFF_CVAR  = OFF_HVAR  + SZ_HV;
constexpr size_t OFF_HPREV = OFF_CVAR  + SZ_HV;
constexpr size_t OFF_DELTA = OFF_HPREV + SZ_HV;
constexpr size_t OFF_GATES = OFF_DELTA + SZ_HV;                  // 4*SZ_HV
constexpr size_t OFF_MIXG  = OFF_GATES + 4*SZ_HV;                // 4*B*NU
constexpr size_t OFF_HMIX  = OFF_MIXG  + 4*(size_t)BATCH*NUU;
constexpr size_t OFF_CMIX  = OFF_HMIX  + (size_t)BATCH*NUU;
constexpr size_t OFF_HML   = OFF_CMIX  + (size_t)BATCH*NUU;
constexpr size_t OFF_QKV   = OFF_HML   + (size_t)BATCH*NUU;      // B*D*384
constexpr size_t OFF_CTX   = OFF_QKV   + (size_t)BATCH*DIMD*384;
constexpr size_t OFF_Z     = OFF_CTX   + SZ_HV;
constexpr size_t OFF_AT    = OFF_Z     + SZ_HV;                  // B*16*16
constexpr size_t OFF_PHI   = OFF_AT    + (size_t)BATCH*DIMD*DIMD;// B*KP*H
constexpr size_t OFF_A     = OFF_PHI   + (size_t)BATCH*KP*HID;   // B*KP*KP
constexpr size_t OFF_BV    = OFF_A     + (size_t)BATCH*KP*KP;    // B*KP
constexpr size_t OFF_TH    = OFF_BV    + (size_t)BATCH*KP;       // B*KP
constexpr size_t OFF_WT    = OFF_TH    + (size_t)BATCH*KP;       // 4*D*H*H (transposed var weights)
constexpr size_t OFF_WMT   = OFF_WT    + (size_t)4*DIMD*HID*HID; // 4*NU*NU (transposed mix weights)

// ---------------- WMMA helpers (V_WMMA_F32_16X16X4_F32) ----------------
typedef float v2f __attribute__((ext_vector_type(2)));
typedef float v8f __attribute__((ext_vector_type(8)));

__device__ inline v8f wmma_f32_4(v2f a, v2f b, v8f c) {
  // D(16x16,f32) = A(16x4,f32) * B(4x16,f32) + C
  return __builtin_amdgcn_wmma_f32_16x16x4_f32(
      false, a, false, b, (short)0, c, false, false);
}

// A-frag (16x4 f32): lane l -> m=l&15 ; holds A[m, 2*(l>>4)] , A[m, 2*(l>>4)+1]
// B-frag (4x16 f32): lane l -> n=l&15 ; holds B[2*(l>>4), n] , B[2*(l>>4)+1, n]
// C/D  (16x16 f32): VGPR r -> row r + 8*(l>>4), col l&15

// CDNA5 async DMA: global -> LDS, tracked by ASYNCcnt.
// NOTE: the LDS element pointer is passed and cast here, so the shared array's
// address ESCAPES into the asm -> compiler must honor the "memory" clobber and
// keep the subsequent ds_load reads of the staged data.
__device__ inline void async_ld_b128(float* lds_ptr, const float* gptr) {
  unsigned       loff = (unsigned)(unsigned long long)(uintptr_t)lds_ptr; // low 32 = LDS offset
  unsigned long long ga = (unsigned long long)(uintptr_t)gptr;
  asm volatile("global_load_async_to_lds_b128 %0, %1, off"
               :: "v"(loff), "v"(ga) : "memory");
}
__device__ inline void wait_async0() {
  asm volatile("s_wait_asynccnt 0x0" ::: "memory");
}

// One wave computes a 16x64 strip of C = A(16xK) * op(B).
// BT=false: B is (K x *) row-major; BT=true: B is stored (N x K) row-major (NT gemm).
template<bool BT>
__device__ inline void wmma_strip(const float* __restrict__ A, int lda,
                                  const float* __restrict__ Bm, int ldb,
                                  int Kdim, v8f acc[4]) {
  const int lane = threadIdx.x & 31;
  const int hf   = lane >> 4;
  const int l16  = lane & 15;
  for (int k0 = 0; k0 < Kdim; k0 += 4) {
    const int ka = k0 + 2 * hf;
    v2f a;
    a.x = A[(size_t)l16 * lda + ka];
    a.y = A[(size_t)l16 * lda + ka + 1];
#pragma unroll
    for (int tc = 0; tc < 4; ++tc) {
      const int n = tc * 16 + l16;
      v2f bv;
      if (BT) { bv.x = Bm[(size_t)n * ldb + ka];  bv.y = Bm[(size_t)n * ldb + ka + 1]; }
      else    { bv.x = Bm[(size_t)ka * ldb + n];  bv.y = Bm[(size_t)(ka + 1) * ldb + n]; }
      acc[tc] = wmma_f32_4(a, bv, acc[tc]);
    }
  }
}

__device__ inline void store_strip(float* __restrict__ C, int ldc, const v8f acc[4]) {
  const int lane = threadIdx.x & 31;
  const int hf   = lane >> 4;
  const int l16  = lane & 15;
#pragma unroll
  for (int tc = 0; tc < 4; ++tc)
#pragma unroll
    for (int r = 0; r < 8; ++r)
      C[(size_t)(r + 8 * hf) * ldc + tc * 16 + l16] = acc[tc][r];
}

__device__ inline float sigf(float x) { return 1.0f / (1.0f + expf(-x)); }

__device__ inline void pair_ij(int p, int& i, int& j) {
  i = p / 15;
  int jj = p - i * 15;
  j = jj + (jj >= i ? 1 : 0);
}

// ---------------- utility kernels ----------------
__global__ void k_zero(float* __restrict__ p, int n) {
  int i = blockIdx.x * blockDim.x + threadIdx.x;
  if (i < n) p[i] = 0.0f;
}
__global__ void k_copy(float* __restrict__ dst, const float* __restrict__ src, int n) {
  int i = blockIdx.x * blockDim.x + threadIdx.x;
  if (i < n) dst[i] = src[i];
}

// ---------------- one-time weight transposes (NN -> NT layout) ----------------
__global__ __launch_bounds__(256) void k_wtrans_var(
    const float* __restrict__ Wj, const float* __restrict__ Wi,
    const float* __restrict__ Wf, const float* __restrict__ Wo,
    float* __restrict__ WT) {
  const int idx = blockIdx.x * blockDim.x + threadIdx.x;   // 4*D*H*H
  const int g = idx >> 18;                                  // D*H*H = 2^18
  const int rem = idx & ((1 << 18) - 1);
  const int d = rem >> 14;                                  // H*H = 2^14
  const int nk = rem & 16383;
  const int n = nk >> 7, k = nk & 127;
  const float* W = (g == 0) ? Wj : (g == 1) ? Wi : (g == 2) ? Wf : Wo;
  WT[idx] = W[(size_t)d * HID * HID + (size_t)k * HID + n];
}
__global__ __launch_bounds__(256) void k_wtrans_mix(
    const float* __restrict__ wj, const float* __restrict__ wi,
    const float* __restrict__ wf, const float* __restrict__ wo,
    float* __restrict__ wT) {
  const int idx = blockIdx.x * blockDim.x + threadIdx.x;   // 4*NU*NU
  const int g = idx >> 14;
  const int nk = idx & 16383;
  const int n = nk >> 7, k = nk & 127;
  const float* W = (g == 0) ? wj : (g == 1) ? wi : (g == 2) ? wf : wo;
  wT[idx] = W[(size_t)k * NUU + n];
}

// ---------------- generic 64x128 NT GEMM block (8 waves) ----------------
template<bool BT>
__global__ __launch_bounds__(256) void k_gemm_generic(
    const float* __restrict__ A, int lda, const float* __restrict__ Bm, int ldb,
    float* __restrict__ C, int ldc, int Kdim) {
  const int w = threadIdx.x >> 5;
  const int rs = (w & 3) * 16, cg = (w >> 2) * 64;
  const float* Ab = A + (size_t)(blockIdx.x * 64 + rs) * lda;
  const int coff = blockIdx.y * 128 + cg;
  const float* Bb = BT ? (Bm + (size_t)coff * ldb) : (Bm + coff);
  __builtin_prefetch(Bb, 0, 1);
  float* Cb = C + (size_t)(blockIdx.x * 64 + rs) * ldc + coff;
  v8f acc[4] = {};
  wmma_strip<BT>(Ab, lda, Bb, ldb, Kdim, acc);
  store_strip(Cb, ldc, acc);
}

// ---------------- var-LSTM gate GEMM (NT against transposed weights) ----------------
__global__ __launch_bounds__(256) void k_vargate(
    const float* __restrict__ hv, const float* __restrict__ WT,
    float* __restrict__ gates) {
  const int g = blockIdx.x, d = blockIdx.y;
  const int w = threadIdx.x >> 5;
  const int rs = (w & 3) * 16, cg = (w >> 2) * 64;
  const int lda = DIMD * HID;
  const float* Bb = WT + ((size_t)g * DIMD + d) * HID * HID + (size_t)cg * HID;
  v8f acc[4] = {};
  wmma_strip<true>(hv + (size_t)d * HID + (size_t)rs * lda, lda, Bb, HID, HID, acc);
  store_strip(gates + (size_t)g * SZ_HV + (size_t)d * HID + (size_t)rs * lda + cg,
              lda, acc);
}

// ---------------- var-LSTM pointwise update ----------------
__global__ __launch_bounds__(256) void k_varpoint(
    const float* __restrict__ gates, const float* __restrict__ x,
    const float* __restrict__ Uj, const float* __restrict__ Ui,
    const float* __restrict__ Uf, const float* __restrict__ Uo,
    const float* __restrict__ Bj, const float* __restrict__ Bi,
    const float* __restrict__ Bf, const float* __restrict__ Bo,
    float* __restrict__ h_var, float* __restrict__ c_var,
    float* __restrict__ h_prev, float* __restrict__ delta, int t) {
  const int idx = blockIdx.x * blockDim.x + threadIdx.x;   // B*D*H
  const int b = idx >> 11, d = (idx >> 7) & 15, h = idx & 127;
  const float xv = x[((size_t)b * SEQT + t) * DIMD + d];
  const int dh = d * HID + h;
  const int S = (int)SZ_HV;
  float pj = gates[idx]         + xv * Uj[dh] + Bj[dh];
  float pi = gates[S + idx]     + xv * Ui[dh] + Bi[dh];
  float pf = gates[2 * S + idx] + xv * Uf[dh] + Bf[dh];
  float po = gates[3 * S + idx] + xv * Uo[dh] + Bo[dh];
  float jv = tanhf(pj), iv = sigf(pi), fv = sigf(pf), ov = sigf(po);
  float c = c_var[idx] * fv + iv * jv;
  c_var[idx] = c;
  float hn = ov * tanhf(c);
  float ho = h_var[idx];
  h_prev[idx] = ho;
  delta[idx]  = hn - ho;
  h_var[idx]  = hn;
}

// ---------------- mix-LSTM GEMMs (NT against transposed weights) ----------------
__global__ __launch_bounds__(256) void k_mixgemm(
    const float* __restrict__ hmix, const float* __restrict__ wT,
    float* __restrict__ mixg) {
  const int g = blockIdx.x;
  const int w = threadIdx.x >> 5;
  const int rs = (w & 3) * 16, cg = (w >> 2) * 64;
  const float* Bb = wT + (size_t)g * NUU * NUU + (size_t)cg * NUU;
  v8f acc[4] = {};
  wmma_strip<true>(hmix + (size_t)rs * NUU, NUU, Bb, NUU, NUU, acc);
  store_strip(mixg + (size_t)g * BATCH * NUU + (size_t)rs * NUU + cg, NUU, acc);
}

// ---------------- mix-LSTM pointwise ----------------
__global__ __launch_bounds__(256) void k_mixpoint(
    const float* __restrict__ mixg, const float* __restrict__ x,
    const float* __restrict__ uj, const float* __restrict__ ui,
    const float* __restrict__ uf, const float* __restrict__ uo,
    const float* __restrict__ bj, const float* __restrict__ bi,
    const float* __restrict__ bf, const float* __restrict__ bo,
    float* __restrict__ c_mix, float* __restrict__ hml, int t) {
  const int idx = blockIdx.x * blockDim.x + threadIdx.x;   // B*NU
  const int b = idx >> 7, n = idx & 127;
  float aj = 0.f, ai = 0.f, af = 0.f, ao = 0.f;
#pragma unroll
  for (int d = 0; d < DIMD; ++d) {
    float xv = x[((size_t)b * SEQT + t) * DIMD + d];
    aj += xv * uj[d * NUU + n];
    ai += xv * ui[d * NUU + n];
    af += xv * uf[d * NUU + n];
    ao += xv * uo[d * NUU + n];
  }
  const int S = BATCH * NUU;
  float jv = tanhf(mixg[idx]        + aj + bj[n]);
  float iv = sigf (mixg[S + idx]    + ai + bi[n]);
  float fv = sigf (mixg[2*S + idx]  + af + bf[n]);
  float ov = sigf (mixg[3*S + idx]  + ao + bo[n]);
  float c = c_mix[idx] * fv + iv * jv;
  c_mix[idx] = c;
  hml[idx] = ov * tanhf(c);
}

// ---------------- attention (16x16 per head, scalar) ----------------
__global__ __launch_bounds__(256) void k_attn(
    const float* __restrict__ qkv, float* __restrict__ ctx, float* __restrict__ a_t) {
  const int b = blockIdx.x, tid = threadIdx.x;
  __shared__ float sattn[NHEAD][16][16];
  const float* qb = qkv + (size_t)b * DIMD * 384;
  if (tid < 128) {
    const int head = tid >> 4, dq = tid & 15;
    float qv[16], s[16];
#pragma unroll
    for (int e = 0; e < 16; ++e) qv[e] = qb[dq * 384 + head * 16 + e];
    float mx = -1e30f;
#pragma unroll
    for (int kk = 0; kk < 16; ++kk) {
      float acc = 0.f;
#pragma unroll
      for (int e = 0; e < 16; ++e) acc += qv[e] * qb[kk * 384 + 128 + head * 16 + e];
      s[kk] = acc * 0.25f;                  // /sqrt(16)
      mx = fmaxf(mx, s[kk]);
    }
    float sum = 0.f;
#pragma unroll
    for (int kk = 0; kk < 16; ++kk) { s[kk] = expf(s[kk] - mx); sum += s[kk]; }
    const float inv = 1.0f / sum;
#pragma unroll
    for (int kk = 0; kk < 16; ++kk) {
      float aa = s[kk] * inv;
      aa = (aa >= THRv) ? aa : 0.0f;
      s[kk] = aa;
      sattn[head][dq][kk] = aa;
    }
    float cv[16];
#pragma unroll
    for (int e = 0; e < 16; ++e) cv[e] = 0.f;
#pragma unroll
    for (int kk = 0; kk < 16; ++kk)
#pragma unroll
      for (int e = 0; e < 16; ++e) cv[e] += s[kk] * qb[kk * 384 + 256 + head * 16 + e];
#pragma unroll
    for (int e = 0; e < 16; ++e)
      ctx[((size_t)b * DIMD + dq) * HID + head * 16 + e] = cv[e];
  }
  __syncthreads();
  const int dq2 = tid >> 4, kk2 = tid & 15;
  float acc = 0.f;
#pragma unroll
  for (int hh = 0; hh < NHEAD; ++hh) acc += sattn[hh][dq2][kk2];
  a_t[(size_t)b * 256 + dq2 * 16 + kk2] = acc * (1.0f / NHEAD);
}

// ---------------- PHI rows 0..31 ----------------
__global__ __launch_bounds__(256) void k_phi_head(
    const float* __restrict__ h_prev, const float* __restrict__ delta,
    float* __restrict__ PHI) {
  const int idx = blockIdx.x * blockDim.x + threadIdx.x;   // B*32*H
  const int b = idx >> 12, r = (idx >> 7) & 31, h = idx & 127;
  float v = (r < 16) ? h_prev[((size_t)b * DIMD + r) * HID + h]
                     : delta [((size_t)b * DIMD + (r - 16)) * HID + h];
  PHI[(size_t)b * KP * HID + (size_t)r * HID + h] = v;
}

// ---------------- psi GEMM: W_psi staged in LDS via async DMA ----------------
__global__ __launch_bounds__(192) void k_psi(
    const float* __restrict__ z, const float* __restrict__ Wpsi,
    const float* __restrict__ bpsi, const float* __restrict__ a_t,
    float* __restrict__ PHI) {
  const int b = blockIdx.y;
  const int tid = threadIdx.x;
  const int w = tid >> 5;
  const int lane = tid & 31;
  const int hf = lane >> 4, l16 = lane & 15;
  const int rs = (w % 3) * 16, cg = (w / 3) * 64;
  const int p0 = blockIdx.x * 48 + rs;                 // 5 blocks x 48 rows = 240

  __shared__ __align__(16) float ldsW[HID * HID];      // 64 KB, reused by 6 waves
  for (int c = tid; c < HID * 32; c += 192) {          // 128 rows x 32 b128 chunks
    const int row = c >> 5, kc = (c & 31) << 2;
    async_ld_b128(&ldsW[row * HID + kc], Wpsi + (size_t)row * HID + kc);
  }
  wait_async0();
  __syncthreads();

  int ii, jj;
  pair_ij(p0 + l16, ii, jj);
  const float* rI = z + ((size_t)b * DIMD + ii) * HID;
  const float* rJ = z + ((size_t)b * DIMD + jj) * HID;
  v8f acc[4] = {};
  for (int k0 = 0; k0 < HID; k0 += 4) {
    const int ka = k0 + 2 * hf;
    v2f a;
    a.x = rI[ka] * rJ[ka];
    a.y = rI[ka + 1] * rJ[ka + 1];
#pragma unroll
    for (int tc = 0; tc < 4; ++tc) {
      const int n = cg + tc * 16 + l16;
      v2f bv;
      bv.x = ldsW[n * HID + ka];
      bv.y = ldsW[n * HID + ka + 1];
      acc[tc] = wmma_f32_4(a, bv, acc[tc]);
    }
  }
  float* Pb = PHI + (size_t)b * KP * HID;
#pragma unroll
  for (int r = 0; r < 8; ++r) {
    const int pr = p0 + r + 8 * hf;
    int pi, pj;
    pair_ij(pr, pi, pj);
    const float wv = a_t[(size_t)b * 256 + pi * 16 + pj];
#pragma unroll
    for (int tc = 0; tc < 4; ++tc) {
      const int col = cg + tc * 16 + l16;
      Pb[(size_t)(32 + pr) * HID + col] = tanhf(acc[tc][r] + bpsi[col]) * wv;
    }
  }
}

// ---------------- Gram: A[b] = PHI_b * PHI_b^T + LAM*I ----------------
// B panel async-staged into LDS (padded rows), shared by 4 waves.
__global__ __launch_bounds__(128) void k_gram(const float* __restrict__ PHI,
                                              float* __restrict__ Ag) {
  const int b = blockIdx.z;
  const int brow = blockIdx.x * 64, bcol = blockIdx.y * 64;
  const int tid = threadIdx.x;
  const int w = tid >> 5;
  const int lane = tid & 31;
  const int hf = lane >> 4, l16 = lane & 15;
  const float* Pb = PHI + (size_t)b * KP * HID;

  constexpr int LSTR = HID + 4;                        // pad (528B rows, 16B aligned)
  __shared__ __align__(16) float ldsB[64 * LSTR];
  for (int c = tid; c < 64 * 32; c += 128) {           // 64 rows x 32 b128 chunks
    const int row = c >> 5, kc = (c & 31) << 2;
    async_ld_b128(&ldsB[row * LSTR + kc], Pb + (size_t)(bcol + row) * HID + kc);
  }
  wait_async0();
  __syncthreads();

  const float* Arow = Pb + (size_t)(brow + w * 16) * HID;
  v8f acc[4] = {};
  for (int k0 = 0; k0 < HID; k0 += 4) {
    const int ka = k0 + 2 * hf;
    v2f a;
    a.x = Arow[(size_t)l16 * HID + ka];
    a.y = Arow[(size_t)l16 * HID + ka + 1];
#pragma unroll
    for (int tc = 0; tc < 4; ++tc) {
      const int n = tc * 16 + l16;
      v2f bv;
      bv.x = ldsB[n * LSTR + ka];
      bv.y = ldsB[n * LSTR + ka + 1];
      acc[tc] = wmma_f32_4(a, bv, acc[tc]);
    }
  }
  float* Cb = Ag + (size_t)b * KP * KP + (size_t)(brow + w * 16) * KP + bcol;
#pragma unroll
  for (int tc = 0; tc < 4; ++tc)
#pragma unroll
    for (int r = 0; r < 8; ++r) {
      const int gr = brow + w * 16 + r + 8 * hf;
      const int gc = bcol + tc * 16 + l16;
      Cb[(size_t)(r + 8 * hf) * KP + tc * 16 + l16] =
          acc[tc][r] + ((gr == gc) ? LAMv : 0.0f);
    }
}

// ---------------- bvec[b,k] = sum_h PHI[b,k,h] * hml[b,h] ----------------
__global__ __launch_bounds__(256) void k_bvec(const float* __restrict__ PHI,
                                              const float* __restrict__ hml,
                                              float* __restrict__ bvec) {
  const int idx = blockIdx.x * blockDim.x + threadIdx.x;   // B*KP
  const int b = idx / KP, k = idx % KP;
  const float* pr = PHI + ((size_t)b * KP + k) * HID;
  const float* hm = hml + (size_t)b * NUU;
  float acc = 0.f;
  for (int h = 0; h < HID; ++h) acc += pr[h] * hm[h];
  bvec[idx] = acc;
}

// ---------------- per-batch in-place LU solve (SPD, no pivoting) ----------------
__global__ __launch_bounds__(256) void k_solve(float* __restrict__ Ag,
                                               float* __restrict__ bv,
                                               float* __restrict__ th) {
  const int b = blockIdx.x, tid = threadIdx.x, nt = blockDim.x;
  float* Ab = Ag + (size_t)b * KP * KP;
  float* bb = bv + (size_t)b * KP;
  float* tb = th + (size_t)b * KP;
  __shared__ float prow[KK];
  __shared__ float sc[2];
  // forward elimination
  for (int p = 0; p < KK - 1; ++p) {
    for (int c = tid; c < KK - p; c += nt) prow[c] = Ab[(size_t)p * KP + p + c];
    if (tid == 0) sc[0] = bb[p];
    __syncthreads();
    const float pinv = 1.0f / prow[0];
    const float pb = sc[0];
    for (int r = p + 1 + tid; r < KK; r += nt) {
      float* Ar = Ab + (size_t)r * KP;
      const float l = Ar[p] * pinv;
      for (int c = p + 1; c < KK; ++c) Ar[c] -= l * prow[c - p];
      bb[r] -= l * pb;
    }
    __syncthreads();
  }
  // back substitution
  for (int p = KK - 1; p >= 0; --p) {
    if (tid == 0) {
      const float tp = bb[p] / Ab[(size_t)p * KP + p];
      sc[1] = tp;
      tb[p] = tp;
    }
    __syncthreads();
    const float tp = sc[1];
    for (int r = tid; r < p; r += nt) bb[r] -= Ab[(size_t)r * KP + p] * tp;
    __syncthreads();
  }
}

// ---------------- h_hat / pred / outputs / h_mix update (t >= 3 only) ----------------
__global__ __launch_bounds__(128) void k_output(
    const float* __restrict__ PHI, const float* __restrict__ theta,
    const float* __restrict__ w_p, const float* __restrict__ b_p,
    float* __restrict__ h_mix, float* __restrict__ out, int t) {
  const int b = blockIdx.x, h = threadIdx.x;     // 128 threads
  const float* pb = PHI + (size_t)b * KP * HID;
  const float* th = theta + (size_t)b * KP;
  float acc = 0.f;
  for (int k = 0; k < KK; ++k) acc += pb[(size_t)k * HID + h] * th[k];
  __shared__ float sh[128];
  sh[h] = acc * w_p[h];
  __syncthreads();
  for (int s = 64; s > 0; s >>= 1) {
    if (h < s) sh[h] += sh[h + s];
    __syncthreads();
  }
  h_mix[(size_t)b * NUU + h] = acc;              // t > SHORT guaranteed here
  const int ti = t - TOFF;
  float* pred_base  = out;
  float* alpha_base = out + (size_t)BATCH * 44;
  float* beta_base  = alpha_base + (size_t)44 * BATCH * DIMD;
  float* gamma_base = beta_base  + (size_t)44 * BATCH * DIMD;
  if (h == 0) pred_base[(size_t)b * 44 + ti] = sh[0] + b_p[0];
  if (h < 16) {
    alpha_base[((size_t)ti * BATCH + b) * DIMD + h] = th[h];
    beta_base [((size_t)ti * BATCH + b) * DIMD + h] = th[16 + h];
  }
  for (int q = h; q < 256; q += 128) {
    const int i = q >> 4, j = q & 15;
    float v = 0.f;
    if (i != j) {
      const int jj = (j < i) ? j : j - 1;
      v = th[32 + i * 15 + jj];
    }
    gamma_base[((size_t)ti * BATCH + b) * 256 + q] = v;
  }
}

// ---------------- host orchestration ----------------
extern "C" void kernel_launch(void* const* d_in, const int* in_sizes, int n_in,
                              void* d_out, int out_size, void* d_ws, size_t ws_size,
                              hipStream_t stream) {
  (void)in_sizes; (void)n_in; (void)out_size; (void)ws_size;
  const float* x     = (const float*)d_in[0];
  const float* W_j   = (const float*)d_in[1];
  const float* W_i   = (const float*)d_in[2];
  const float* W_f   = (const float*)d_in[3];
  const float* W_o   = (const float*)d_in[4];
  const float* U_j   = (const float*)d_in[5];
  const float* U_i   = (const float*)d_in[6];
  const float* U_f   = (const float*)d_in[7];
  const float* U_o   = (const float*)d_in[8];
  const float* B_j   = (const float*)d_in[9];
  const float* B_i   = (const float*)d_in[10];
  const float* B_f   = (const float*)d_in[11];
  const float* B_o   = (const float*)d_in[12];
  const float* u_j   = (const float*)d_in[13];
  const float* u_i   = (const float*)d_in[14];
  const float* u_f   = (const float*)d_in[15];
  const float* u_o   = (const float*)d_in[16];
  const float* w_j   = (const float*)d_in[17];
  const float* w_i   = (const float*)d_in[18];
  const float* w_f   = (const float*)d_in[19];
  const float* w_o   = (const float*)d_in[20];
  const float* b_j   = (const float*)d_in[21];
  const float* b_i   = (const float*)d_in[22];
  const float* b_f   = (const float*)d_in[23];
  const float* b_o   = (const float*)d_in[24];
  const float* W_qkv = (const float*)d_in[25];
  const float* W_out = (const float*)d_in[26];
  const float* W_psi = (const float*)d_in[27];
  const float* b_psi = (const float*)d_in[28];
  const float* w_p   = (const float*)d_in[29];
  const float* b_p   = (const float*)d_in[30];

  float* ws     = (float*)d_ws;
  float* h_var  = ws + OFF_HVAR;
  float* c_var  = ws + OFF_CVAR;
  float* h_prev = ws + OFF_HPREV;
  float* delta  = ws + OFF_DELTA;
  float* gates  = ws + OFF_GATES;
  float* mixg   = ws + OFF_MIXG;
  float* h_mix  = ws + OFF_HMIX;
  float* c_mix  = ws + OFF_CMIX;
  float* hml    = ws + OFF_HML;
  float* qkv    = ws + OFF_QKV;
  float* ctx    = ws + OFF_CTX;
  float* z      = ws + OFF_Z;
  float* a_t    = ws + OFF_AT;
  float* PHI    = ws + OFF_PHI;
  float* Agram  = ws + OFF_A;
  float* bvec   = ws + OFF_BV;
  float* theta  = ws + OFF_TH;
  float* WT     = ws + OFF_WT;
  float* wMT    = ws + OFF_WMT;
  float* out    = (float*)d_out;

  // zero initial state + PHI (padding rows must stay zero forever)
  k_zero<<<(int)(2 * SZ_HV + 255) / 256, 256, 0, stream>>>(h_var, (int)(2 * SZ_HV));
  k_zero<<<(2 * BATCH * NUU + 255) / 256, 256, 0, stream>>>(h_mix, 2 * BATCH * NUU);
  {
    const int nphi = BATCH * KP * HID;
    k_zero<<<(nphi + 255) / 256, 256, 0, stream>>>(PHI, nphi);
  }
  // one-time weight transposes into NT layout (fused b64 B-fragment loads)
  k_wtrans_var<<<(4 * DIMD * HID * HID) / 256, 256, 0, stream>>>(W_j, W_i, W_f, W_o, WT);
  k_wtrans_mix<<<(4 * NUU * NUU) / 256, 256, 0, stream>>>(w_j, w_i, w_f, w_o, wMT);

  for (int t = 0; t < TSTEPS; ++t) {
    k_vargate<<<dim3(4, DIMD), 256, 0, stream>>>(h_var, WT, gates);
    k_varpoint<<<(int)(SZ_HV / 256), 256, 0, stream>>>(
        gates, x, U_j, U_i, U_f, U_o, B_j, B_i, B_f, B_o,
        h_var, c_var, h_prev, delta, t);
    k_mixgemm<<<4, 256, 0, stream>>>(h_mix, wMT, mixg);
    k_mixpoint<<<(BATCH * NUU) / 256, 256, 0, stream>>>(
        mixg, x, u_j, u_i, u_f, u_o, b_j, b_i, b_f, b_o, c_mix, hml, t);

    if (t > 2) {
      // qkv = h_var @ W_qkv^T : (1024 x 128) x (128 x 384), NT
      k_gemm_generic<true><<<dim3(16, 3), 256, 0, stream>>>(
          h_var, HID, W_qkv, HID, qkv, 384, HID);
      k_attn<<<BATCH, 256, 0, stream>>>(qkv, ctx, a_t);
      // z = ctx @ W_out^T
      k_gemm_generic<true><<<dim3(16, 1), 256, 0, stream>>>(
          ctx, HID, W_out, HID, z, HID, HID);
      k_phi_head<<<(BATCH * 32 * HID) / 256, 256, 0, stream>>>(h_prev, delta, PHI);
      k_psi<<<dim3(5, BATCH), 192, 0, stream>>>(z, W_psi, b_psi, a_t, PHI);
      k_gram<<<dim3(KP / 64, KP / 64, BATCH), 128, 0, stream>>>(PHI, Agram);
      k_bvec<<<(BATCH * KP) / 256, 256, 0, stream>>>(PHI, hml, bvec);
      k_solve<<<BATCH, 256, 0, stream>>>(Agram, bvec, theta);
      k_output<<<BATCH, 128, 0, stream>>>(PHI, theta, w_p, b_p, h_mix, out, t);
    } else {
      k_copy<<<(BATCH * NUU) / 256, 256, 0, stream>>>(h_mix, hml, BATCH * NUU);
    }
  }
}